// RayAlignedLoss_34076270526589
// MI455X (gfx1250) — compile-verified
//
#include <hip/hip_runtime.h>
#include <hip/hip_bf16.h>
#include <math.h>

// ---------------------------------------------------------------------------
// RayAlignedLoss for MI455X (gfx1250, wave32)
//
// Pass 1: per-voxel logsumexp over 18 channels, channel reduction done with
//         chained V_WMMA_F32_16X16X4_F32 (16 voxels/wave, K padded 18->20).
// Pass 2: one thread per ray, early-exit march, gathers lse + 1 logit/step.
// Pass 3: finalize scalar loss.
// ---------------------------------------------------------------------------

typedef float v2f __attribute__((ext_vector_type(2)));
typedef float v8f __attribute__((ext_vector_type(8)));

#define NVOX_PER_B   640000          // 200*200*16
#define NVOX_TOTAL   1280000         // B=2
#define NCH          18
#define FREE_CH      17
#define NUM_RAYS     8000
#define NEG_LOG_EPS  13.815510558f   // -log(1e-6)
#define PI_F         3.14159265358979323846f

// ---------------- Pass 1: lse[b*640000+flat] = log(sum_c exp(logit_c)) -----
// One wave handles 16 consecutive voxels. A-matrix (16x4 f32) layout:
//   lanes 0-15  : M=lane, K = {chunk*4+0, chunk*4+1}
//   lanes 16-31 : M=lane-16, K = {chunk*4+2, chunk*4+3}
// B = all-ones => every column of D = row sums; lanes 0/16 hold M=0..7 / 8..15
// in their 8 accumulator VGPRs and write the results.
__global__ __launch_bounds__(256)
void lse_wmma_kernel(const float* __restrict__ logits, float* __restrict__ lse) {
    const int wave  = blockIdx.x * (blockDim.x >> 5) + (threadIdx.x >> 5);
    const int lane  = threadIdx.x & 31;
    const int vbase = wave * 16;                 // 16 voxels per wave
    if (vbase >= NVOX_TOTAL) return;             // wave-uniform guard (EXEC stays all-ones)

    const int m     = lane & 15;                 // voxel row within tile
    const int khalf = (lane >> 4) * 2;           // 0 for lanes 0-15, 2 for 16-31
    const float* __restrict__ p = logits + (size_t)(vbase + m) * NCH;

    // Prefetch the next tile's logits while we compute this one.
    __builtin_prefetch(p + (size_t)16 * NCH * (blockDim.x >> 5), 0, 1);

    v8f c = {0.f, 0.f, 0.f, 0.f, 0.f, 0.f, 0.f, 0.f};
    v2f bones; bones.x = 1.f; bones.y = 1.f;

#pragma unroll
    for (int chunk = 0; chunk < 5; ++chunk) {    // K = 20 (18 real + 2 pad)
        const int k0 = chunk * 4 + khalf;
        v2f a;
        a.x = (k0     < NCH) ? __expf(p[k0])     : 0.f;
        a.y = (k0 + 1 < NCH) ? __expf(p[k0 + 1]) : 0.f;
        c = __builtin_amdgcn_wmma_f32_16x16x4_f32(false, a, false, bones,
                                                  (short)0, c, false, false);
    }

    // C/D layout: VGPR j -> M=j (lanes 0-15), M=j+8 (lanes 16-31).
    if (lane == 0 || lane == 16) {
        const int off = (lane == 0) ? 0 : 8;
        float* __restrict__ dst = lse + vbase + off;
#pragma unroll
        for (int j = 0; j < 8; ++j)
            dst[j] = __logf(c[j]);
    }
}

// ---------------- ray table generation (host constants re-derived) ---------
// Reference builds ~39 pitch rows x 360 azimuths, then RandomState(0).choice
// picks 8000 without replacement (Mersenne Twister - not reproducible in a
// kernel). We take a deterministic coprime-stride 8000-subset instead.
__global__ void gen_rays_kernel(float4* __restrict__ rays) {
    const int tid = blockIdx.x * blockDim.x + threadIdx.x;
    if (tid >= NUM_RAYS) return;

    float pitch[64];
    int np = 0;
    for (int k = 0; k < 10; ++k)
        pitch[np++] = -(PI_F * 0.5f - atanf((float)(k + 1)));
    while (pitch[np - 1] < 0.21f && np < 64) {
        const float d = pitch[np - 1] - pitch[np - 2];
        pitch[np] = pitch[np - 1] + d;
        ++np;
    }
    const int total = np * 360;
    const int idx   = (int)(((long long)tid * 7919) % total);   // gcd(7919,total)=1

    const float pt = pitch[idx / 360];
    const float az = (float)(idx % 360) * (PI_F / 180.0f);
    const float cp = cosf(pt);
    float dx = cp * cosf(az), dy = cp * sinf(az), dz = sinf(pt);
    const float ma = fmaxf(fabsf(dx), fmaxf(fabsf(dy), fabsf(dz))) + 1e-8f;
    float4 r; r.x = dx; r.y = dy; r.z = dz; r.w = 0.4f / ma;    // w = dt
    rays[tid] = r;
}

// ---------------- accumulator zeroing (graph-capture safe) -----------------
__global__ void zero_acc_kernel(float* __restrict__ acc) {
    const int tid = blockIdx.x * blockDim.x + threadIdx.x;
    if (tid < 16) acc[tid] = 0.f;
}

// ---------------- Pass 2: ray march ----------------------------------------
// acc layout per batch b: [b*5 + 0]=sum_pre [1]=n_pre [2]=sum_bce_hit
//                         [3]=sum_ce_hit    [4]=n_rays
__global__ __launch_bounds__(256)
void ray_march_kernel(const float*  __restrict__ logits,
                      const float*  __restrict__ origins,
                      const int*    __restrict__ sem,
                      const int*    __restrict__ cam,
                      const float*  __restrict__ lse,
                      const float4* __restrict__ rays,
                      float*        __restrict__ acc) {
    const int tid = blockIdx.x * blockDim.x + threadIdx.x;
    if (tid >= 2 * NUM_RAYS) return;
    const int b = tid / NUM_RAYS;
    const int n = tid - b * NUM_RAYS;

    const float ox = origins[b * 3 + 0];
    const float oy = origins[b * 3 + 1];
    const float oz = origins[b * 3 + 2];
    const float4 r = rays[n];
    const float dt = r.w;

    const size_t vb = (size_t)b * NVOX_PER_B;

    float sum_pre = 0.f;
    float cnt_pre = 0.f;
    int   hit_flat = -1, hit_g = 0;

    for (int s = 0; s < 512; ++s) {
        const float t = dt * ((float)s + 0.5f);
        if (t > 60.0f) break;                              // == t<=MAX_DIST mask
        const float px = fmaf(r.x, t, ox);
        const float py = fmaf(r.y, t, oy);
        const float pz = fmaf(r.z, t, oz);
        const int vx = (int)floorf((px + 40.0f) * 2.5f);   // /VOXEL(0.4)
        const int vy = (int)floorf((py + 40.0f) * 2.5f);
        const int vz = (int)floorf((pz +  1.0f) * 2.5f);
        if ((unsigned)vx >= 200u || (unsigned)vy >= 200u || (unsigned)vz >= 16u)
            continue;                                      // out-of-bounds step
        const int flat = (vx * 200 + vy) * 16 + vz;
        const int g = sem[vb + flat];
        if (g != FREE_CH) { hit_flat = flat; hit_g = g; break; }  // first_hit
        // pre-hit free step: bce_pre = min(-lp_free, -log(EPS))
        const float lpf = logits[(vb + flat) * (size_t)NCH + FREE_CH]
                        - lse[vb + flat];
        sum_pre += fminf(-lpf, NEG_LOG_EPS);
        cnt_pre += 1.f;
    }

    if (hit_flat >= 0 && cam[vb + hit_flat] > 0) {         // ray_valid
        const float l = lse[vb + hit_flat];
        const float* lp = logits + (vb + hit_flat) * (size_t)NCH;
        const float lpf     = lp[FREE_CH] - l;
        const float p_hit   = 1.0f - __expf(lpf);
        const float bce_hit = -__logf(fmaxf(p_hit, 1e-6f));
        const float ce_hit  = -(lp[hit_g] - l);
        atomicAdd(&acc[b * 5 + 0], sum_pre);
        atomicAdd(&acc[b * 5 + 1], cnt_pre);
        atomicAdd(&acc[b * 5 + 2], bce_hit);
        atomicAdd(&acc[b * 5 + 3], ce_hit);
        atomicAdd(&acc[b * 5 + 4], 1.0f);
    }
}

// ---------------- Pass 3: finalize -----------------------------------------
__global__ void finalize_kernel(const float* __restrict__ acc,
                                float* __restrict__ out) {
    if (threadIdx.x != 0 || blockIdx.x != 0) return;
    float loss_pre = 0.f, loss_ho = 0.f, loss_hs = 0.f;
    float n_valid_pre = 0.f, n_valid_hit = 0.f;
    for (int b = 0; b < 2; ++b) {
        const float n_pre  = acc[b * 5 + 1];
        const float n_rays = acc[b * 5 + 4];
        if (n_pre > 0.f)  { loss_pre += acc[b * 5 + 0] / fmaxf(n_pre, 1.f);  n_valid_pre += 1.f; }
        if (n_rays > 0.f) {
            loss_ho += acc[b * 5 + 2] / fmaxf(n_rays, 1.f);
            loss_hs += acc[b * 5 + 3] / fmaxf(n_rays, 1.f);
            n_valid_hit += 1.f;
        }
    }
    loss_pre /= fmaxf(n_valid_pre, 1.f);
    loss_ho  /= fmaxf(n_valid_hit, 1.f);
    loss_hs  /= fmaxf(n_valid_hit, 1.f);
    out[0] = loss_pre + loss_ho + loss_hs;     // W_PRE=W_HIT=W_CE=LOSS_W=1
}

// ---------------------------------------------------------------------------
extern "C" void kernel_launch(void* const* d_in, const int* in_sizes, int n_in,
                              void* d_out, int out_size, void* d_ws, size_t ws_size,
                              hipStream_t stream) {
    const float* occ_logits = (const float*)d_in[0];   // (2,200,200,16,18) f32
    const float* origins    = (const float*)d_in[1];   // (2,3) f32
    const int*   sem        = (const int*)  d_in[2];   // (2,200,200,16) i32
    const int*   cam        = (const int*)  d_in[3];   // (2,200,200,16) i32
    float*       out        = (float*)d_out;

    // workspace layout
    char* ws = (char*)d_ws;
    float*  lse  = (float*) ws;                                   // 1,280,000 f32
    float4* rays = (float4*)(ws + (size_t)NVOX_TOTAL * 4);        // 8000 float4
    float*  acc  = (float*) (ws + (size_t)NVOX_TOTAL * 4
                                + (size_t)NUM_RAYS * 16);         // 16 f32

    zero_acc_kernel<<<1, 32, 0, stream>>>(acc);
    gen_rays_kernel<<<(NUM_RAYS + 255) / 256, 256, 0, stream>>>(rays);

    // 1,280,000 voxels / 16 per wave / 8 waves per block = 10,000 blocks
    lse_wmma_kernel<<<NVOX_TOTAL / (16 * 8), 256, 0, stream>>>(occ_logits, lse);

    ray_march_kernel<<<(2 * NUM_RAYS + 255) / 256, 256, 0, stream>>>(
        occ_logits, origins, sem, cam, lse, rays, acc);

    finalize_kernel<<<1, 32, 0, stream>>>(acc, out);
}